// emdModule_61641370632797
// MI455X (gfx1250) — compile-verified
//
#include <hip/hip_runtime.h>
#include <math.h>

typedef __attribute__((ext_vector_type(2))) float v2f;
typedef __attribute__((ext_vector_type(8))) float v8f;

constexpr int BB = 16;
constexpr int NN = 2048;
constexpr int WAVES = 8;
constexpr int THREADS = WAVES * 32;
#define BIGF 3.402823466e+38f

__global__ __launch_bounds__(THREADS)
void emd_min_assign_kernel(const float* __restrict__ x1,
                           const float* __restrict__ x2,
                           const float* __restrict__ epsp,
                           const int*   __restrict__ itersp,
                           float* __restrict__ outv,
                           int*   __restrict__ outi)
{
    // x2[b] staged once per block: 2048 pts * 3 f32 = 24KB, plus |x2|^2 = 8KB.
    __shared__ float s_x2[NN * 3];
    __shared__ float s_sq2[NN];

    const int tid  = threadIdx.x;
    const int lane = tid & 31;
    const int wave = tid >> 5;
    const int l16  = lane & 15;
    const int half = lane >> 4;          // 0: lanes 0-15, 1: lanes 16-31

    const int tile0 = blockIdx.x * WAVES;     // 128 i-tiles per batch, 8 | 128
    const int b     = tile0 >> 7;             // same batch for all waves in block
    const int tile  = tile0 + wave;
    const int i0    = (tile & 127) << 4;

    const float* x1b = x1 + (size_t)b * NN * 3;
    const float* x2b = x2 + (size_t)b * NN * 3;

    // ---- cooperative LDS staging of x2[b] ----
    for (int k = tid; k < NN * 3; k += THREADS) s_x2[k] = x2b[k];
    __syncthreads();
    for (int j = tid; j < NN; j += THREADS) {
        float px = s_x2[3 * j + 0], py = s_x2[3 * j + 1], pz = s_x2[3 * j + 2];
        s_sq2[j] = px * px + py * py + pz * pz;
    }
    __syncthreads();

    // ---- A fragment: 16x4 f32, K=3 padded to 4, with the distance algebra
    //      folded in:  A[m,0:3] = -2*x1[m,:],  A[m,3] = 1.0
    // lanes 0-15: row m=l16, k={0,1}; lanes 16-31: row m=l16, k={2,3}
    v2f afrag;
    {
        const float* pa = x1b + (size_t)(i0 + l16) * 3;
        afrag.x = -2.0f * (half ? pa[2] : pa[0]);
        afrag.y = half ? 1.0f : (-2.0f * pa[1]);
    }

    // ---- C input: |x1[i]|^2 broadcast along N (loop-invariant accumulator) ----
    // C-layout: VGPR r holds row i0 + r + 8*half for this lane.
    v8f csq1;
#pragma unroll
    for (int r = 0; r < 8; ++r) {
        const float* p = x1b + (size_t)(i0 + r + 8 * half) * 3;
        float qx = p[0], qy = p[1], qz = p[2];
        csq1[r] = qx * qx + qy * qy + qz * qz;
    }

    float minv[8];
    int   minj[8];
#pragma unroll
    for (int r = 0; r < 8; ++r) { minv[r] = BIGF; minj[r] = 0; }

    // ---- sweep 128 j-tiles: one WMMA emits d^2 = |x1|^2 + |x2|^2 - 2<x1,x2>
    //      for a full 16x16 tile; VALU does only clamp + min/argmin. ----
#pragma unroll 2
    for (int jt = 0; jt < NN / 16; ++jt) {
        const int j = (jt << 4) + l16;
        // B fragment: 4x16 f32; lane holds col n=l16.
        //   B[0:3,n] = x2[n,:],  B[3,n] = |x2[n]|^2
        const float px  = s_x2[3 * j + 0];
        const float py  = s_x2[3 * j + 1];
        const float pz  = s_x2[3 * j + 2];
        const float sq2 = s_sq2[j];
        v2f bfrag;
        bfrag.x = half ? pz  : px;
        bfrag.y = half ? sq2 : py;

        v8f c = __builtin_amdgcn_wmma_f32_16x16x4_f32(
            /*neg_a=*/false, afrag, /*neg_b=*/false, bfrag,
            /*c_mod=*/(short)0, csq1, /*reuse_a=*/false, /*reuse_b=*/false);

#pragma unroll
        for (int r = 0; r < 8; ++r) {
            // clamp to [0, +MAX] in one v_med3_f32 (inputs are finite)
            float d2 = __builtin_amdgcn_fmed3f(c[r], 0.0f, BIGF);
            if (d2 < minv[r]) { minv[r] = d2; minj[r] = j; }  // strict <: first j wins
        }
    }

    // ---- closed-form price scan: min_cost = c_min * (1-eps)^(iters-1) ----
    const float eps   = epsp[0];
    const int   iters = itersp[0];
    const float scale = powf(1.0f - eps, (float)(iters - 1));

    // ---- min+argmin butterfly across each 16-lane half-group ----
#pragma unroll
    for (int r = 0; r < 8; ++r) {
        float v = minv[r];
        int   j = minj[r];
#pragma unroll
        for (int m = 8; m >= 1; m >>= 1) {
            float ov = __shfl_xor(v, m, 32);
            int   oj = __shfl_xor(j, m, 32);
            if (ov < v || (ov == v && oj < j)) { v = ov; j = oj; }
        }
        if (l16 == 0) {
            const int i = i0 + r + 8 * half;
            const float cmin = sqrtf(v);                     // min_j ||x1_i - x2_j||
            outv[(size_t)b * NN + i] = sqrtf(cmin * scale);  // sqrt(min_cost)
            outi[(size_t)b * NN + i] = j;                    // argmin (int32)
        }
    }
}

extern "C" void kernel_launch(void* const* d_in, const int* in_sizes, int n_in,
                              void* d_out, int out_size, void* d_ws, size_t ws_size,
                              hipStream_t stream) {
    const float* x1     = (const float*)d_in[0];
    const float* x2     = (const float*)d_in[1];
    const float* epsp   = (const float*)d_in[2];
    const int*   itersp = (const int*)d_in[3];

    float* outv = (float*)d_out;                    // sqrt(min_cost): B*N f32
    int*   outi = (int*)d_out + (size_t)BB * NN;    // assignment:      B*N i32

    const int total_tiles = BB * (NN / 16);         // 2048 waves of work
    dim3 grid(total_tiles / WAVES);                 // 256 blocks x 256 threads
    emd_min_assign_kernel<<<grid, THREADS, 0, stream>>>(x1, x2, epsp, itersp,
                                                        outv, outi);
}